// GraphSAGE_2259152798196
// MI455X (gfx1250) — compile-verified
//
#include <hip/hip_runtime.h>

#define BN_EPS 1e-5f

typedef __attribute__((ext_vector_type(2))) float v2f;
typedef __attribute__((ext_vector_type(8))) float v8f;

// ---------------------------------------------------------------- utilities
__global__ void zero_f32(float* __restrict__ p, long long n) {
  long long i = (long long)blockIdx.x * blockDim.x + threadIdx.x;
  long long stride = (long long)gridDim.x * blockDim.x;
  for (; i < n; i += stride) p[i] = 0.0f;
}

__global__ void copy_to_z(const float* __restrict__ x, float* __restrict__ z,
                          long long total /* N*128 */) {
  long long i = (long long)blockIdx.x * blockDim.x + threadIdx.x;
  if (i >= total) return;
  long long row = i >> 7;
  int col = (int)(i & 127);
  z[row * 512 + col] = x[i];
}

// ---------------------------------------------------------------- degree
__global__ void deg_count(const long long* __restrict__ ei, long long E,
                          float* __restrict__ deg) {
  long long i = (long long)blockIdx.x * blockDim.x + threadIdx.x;
  if (i >= E) return;
  long long d = ei[E + i];  // dst row
  atomicAdd(&deg[d], 1.0f);
}

__global__ void deg_invert(float* __restrict__ deg, long long N) {
  long long i = (long long)blockIdx.x * blockDim.x + threadIdx.x;
  if (i >= N) return;
  float d = deg[i];
  deg[i] = 1.0f / (d > 1.0f ? d : 1.0f);
}

// ------------------------------------------------- gather + scatter-add (mean aggr)
// one thread per (edge, 4-float chunk); zsrc points at the layer's column slice of z
__global__ void scatter_add(const float* __restrict__ zsrc,
                            const long long* __restrict__ ei, long long E,
                            float* __restrict__ agg) {
  long long t = (long long)blockIdx.x * blockDim.x + threadIdx.x;
  long long e = t >> 5;
  if (e >= E) return;
  int ch = (int)((t & 31) << 2);
  long long s = ei[e];
  long long d = ei[E + e];
  const float4 v = *(const float4*)(zsrc + s * 512 + ch);
  float* dp = agg + d * 128 + ch;
  atomicAdd(dp + 0, v.x);
  atomicAdd(dp + 1, v.y);
  atomicAdd(dp + 2, v.z);
  atomicAdd(dp + 3, v.w);
}

__global__ void scale_rows(float* __restrict__ agg,
                           const float* __restrict__ invdeg,
                           long long total /* N*128 */) {
  long long i = (long long)blockIdx.x * blockDim.x + threadIdx.x;
  if (i >= total) return;
  agg[i] *= invdeg[i >> 7];
}

// ---------------------------------------------------------------- fp32 WMMA GEMM
// C[row0:row0+16, col0:col0+16] = bias + A1@W1 (+ A2@W2), K-loop of V_WMMA_F32_16X16X4_F32.
// One wave per 16x16 output tile; blockDim.x = 32 * (ncols/16); gridDim.x = N/16.
// W matrices are row-major [K, ncols] (row stride == ncols).
__global__ void gemm_wmma_f32(const float* __restrict__ A1, int lda1,
                              const float* __restrict__ W1,
                              const float* __restrict__ A2, int lda2,
                              const float* __restrict__ W2,
                              const float* __restrict__ bias,
                              float* __restrict__ C, int ldc,
                              int K, int ncols) {
  const int lane = threadIdx.x & 31;
  const int wave = threadIdx.x >> 5;
  const int mn   = lane & 15;         // M for A-frag, N for B/C-frag
  const int kb   = (lane >> 4) << 1;  // lanes 0-15: K={0,1}; lanes 16-31: K={2,3}
  const long long row0 = (long long)blockIdx.x * 16;
  const int col0 = wave * 16;

  v8f c;
  const float bv = bias[col0 + mn];
#pragma unroll
  for (int r = 0; r < 8; ++r) c[r] = bv;

  {
    const float* ap = A1 + (row0 + mn) * (long long)lda1 + kb;
    const float* wp = W1 + (long long)kb * ncols + col0 + mn;
    for (int k = 0; k < K; k += 4) {
      v2f a; a.x = ap[0]; a.y = ap[1];
      v2f b; b.x = wp[0]; b.y = wp[ncols];
      c = __builtin_amdgcn_wmma_f32_16x16x4_f32(false, a, false, b,
                                                (short)0, c, false, false);
      ap += 4;
      wp += 4 * ncols;
    }
  }
  if (A2 != nullptr) {
    const float* ap = A2 + (row0 + mn) * (long long)lda2 + kb;
    const float* wp = W2 + (long long)kb * ncols + col0 + mn;
    for (int k = 0; k < K; k += 4) {
      v2f a; a.x = ap[0]; a.y = ap[1];
      v2f b; b.x = wp[0]; b.y = wp[ncols];
      c = __builtin_amdgcn_wmma_f32_16x16x4_f32(false, a, false, b,
                                                (short)0, c, false, false);
      ap += 4;
      wp += 4 * ncols;
    }
  }

  // C/D layout: VGPR r, lanes 0-15 -> M=r, lanes 16-31 -> M=r+8; N = lane&15
  const long long rb = row0 + ((lane >> 4) << 3);
#pragma unroll
  for (int r = 0; r < 8; ++r)
    C[(rb + r) * (long long)ldc + col0 + mn] = c[r];
}

// ---------------------------------------------------------------- batchnorm
// blockDim.x == Ccols (one thread per column); each block reduces 512 rows.
__global__ void bn_stats(const float* __restrict__ pre, int Ccols, long long N,
                         float* __restrict__ sum, float* __restrict__ sumsq) {
  const int col = threadIdx.x;
  long long r0 = (long long)blockIdx.x * 512;
  long long r1 = r0 + 512;
  if (r1 > N) r1 = N;
  float s = 0.0f, ss = 0.0f;
  for (long long r = r0; r < r1; ++r) {
    float v = pre[r * Ccols + col];
    s += v;
    ss += v * v;
  }
  atomicAdd(&sum[col], s);
  atomicAdd(&sumsq[col], ss);
}

__global__ void bn_finalize(const float* __restrict__ sum,
                            const float* __restrict__ sumsq,
                            const float* __restrict__ g,
                            const float* __restrict__ b,
                            float* __restrict__ scale,
                            float* __restrict__ shift, long long N) {
  const int c = threadIdx.x;
  float invN = 1.0f / (float)N;
  float mean = sum[c] * invN;
  float var = sumsq[c] * invN - mean * mean;
  float sc = g[c] * rsqrtf(var + BN_EPS);
  scale[c] = sc;
  shift[c] = b[c] - mean * sc;
}

// normalize + activation; dst may be a strided slice (JK concat buffer)
__global__ void bn_apply(const float* __restrict__ pre, long long N,
                         const float* __restrict__ scale,
                         const float* __restrict__ shift,
                         float* __restrict__ dst, int dstStride,
                         float negSlope) {
  long long i = (long long)blockIdx.x * blockDim.x + threadIdx.x;
  long long total = N * 128;
  if (i >= total) return;
  long long row = i >> 7;
  int col = (int)(i & 127);
  float v = pre[i] * scale[col] + shift[col];
  v = (v > 0.0f) ? v : v * negSlope;  // negSlope==0 -> ReLU
  dst[row * (long long)dstStride + col] = v;
}

// ---------------------------------------------------------------- driver
extern "C" void kernel_launch(void* const* d_in, const int* in_sizes, int n_in,
                              void* d_out, int out_size, void* d_ws, size_t ws_size,
                              hipStream_t stream) {
  (void)n_in; (void)out_size; (void)ws_size;

  const float*     x     = (const float*)d_in[0];
  const long long* ei    = (const long long*)d_in[1];
  const float*     Wl    = (const float*)d_in[2];
  const float*     bl    = (const float*)d_in[3];
  const float*     Wr    = (const float*)d_in[4];
  const float*     gamma = (const float*)d_in[5];
  const float*     beta  = (const float*)d_in[6];
  const float*     W1    = (const float*)d_in[7];
  const float*     b1    = (const float*)d_in[8];
  const float*     g1    = (const float*)d_in[9];
  const float*     be1   = (const float*)d_in[10];
  const float*     W2    = (const float*)d_in[11];
  const float*     b2    = (const float*)d_in[12];
  float*           out   = (float*)d_out;

  const long long N = (long long)in_sizes[0] / 128;  // 100000 (divisible by 16)
  const long long E = (long long)in_sizes[1] / 2;    // 1600000
  const int NUM_LAYERS = 3;

  // ---- workspace carve-out
  char* ws = (char*)d_ws;
  size_t off = 0;
  auto carve = [&](size_t bytes) -> float* {
    float* p = (float*)(ws + off);
    off += (bytes + 255) & ~(size_t)255;
    return p;
  };
  float* z      = carve((size_t)N * 512 * 4);  // JK concat [N, 512]
  float* agg    = carve((size_t)N * 128 * 4);  // aggregation / head t1
  float* pre    = carve((size_t)N * 128 * 4);  // pre-BN buffer
  float* invdeg = carve((size_t)N * 4);
  float* ssum   = carve(128 * 4);
  float* ssq    = carve(128 * 4);
  float* bsc    = carve(128 * 4);
  float* bsh    = carve(128 * 4);

  const long long NF = N * 128;  // elements in an [N,128] matrix
  const int rowTiles = (int)(N / 16);

  // ---- degrees -> inv_deg
  zero_f32<<<1024, 256, 0, stream>>>(invdeg, N);
  deg_count<<<(unsigned)((E + 255) / 256), 256, 0, stream>>>(ei, E, invdeg);
  deg_invert<<<(unsigned)((N + 255) / 256), 256, 0, stream>>>(invdeg, N);

  // ---- x into z[:, 0:128]
  copy_to_z<<<(unsigned)((NF + 255) / 256), 256, 0, stream>>>(x, z, NF);

  // ---- SAGE layers
  for (int i = 0; i < NUM_LAYERS; ++i) {
    const float* h = z + i * 128;  // current features: z columns [i*128, i*128+128)
    zero_f32<<<4096, 256, 0, stream>>>(agg, NF);
    scatter_add<<<(unsigned)((E * 32 + 255) / 256), 256, 0, stream>>>(h, ei, E, agg);
    scale_rows<<<(unsigned)((NF + 255) / 256), 256, 0, stream>>>(agg, invdeg, NF);

    // pre = agg @ Wl[i] + bl[i] + h @ Wr[i]   (fused double-WMMA)
    gemm_wmma_f32<<<rowTiles, 256, 0, stream>>>(
        agg, 128, Wl + (size_t)i * 128 * 128,
        h, 512, Wr + (size_t)i * 128 * 128,
        bl + (size_t)i * 128, pre, 128, 128, 128);

    zero_f32<<<1, 256, 0, stream>>>(ssum, 128);
    zero_f32<<<1, 256, 0, stream>>>(ssq, 128);
    bn_stats<<<(unsigned)((N + 511) / 512), 128, 0, stream>>>(pre, 128, N, ssum, ssq);
    bn_finalize<<<1, 128, 0, stream>>>(ssum, ssq, gamma + (size_t)i * 128,
                                       beta + (size_t)i * 128, bsc, bsh, N);
    // ReLU result -> z[:, (i+1)*128 : (i+2)*128]
    bn_apply<<<(unsigned)((NF + 255) / 256), 256, 0, stream>>>(
        pre, N, bsc, bsh, z + (i + 1) * 128, 512, 0.0f);
  }

  // ---- head: pre = z @ W1 + b1   (K=512)
  gemm_wmma_f32<<<rowTiles, 256, 0, stream>>>(
      z, 512, W1, nullptr, 0, nullptr, b1, pre, 128, 512, 128);

  zero_f32<<<1, 256, 0, stream>>>(ssum, 128);
  zero_f32<<<1, 256, 0, stream>>>(ssq, 128);
  bn_stats<<<(unsigned)((N + 511) / 512), 128, 0, stream>>>(pre, 128, N, ssum, ssq);
  bn_finalize<<<1, 128, 0, stream>>>(ssum, ssq, g1, be1, bsc, bsh, N);
  // leaky_relu(0.01) -> t1 (reuse agg buffer, contiguous [N,128])
  bn_apply<<<(unsigned)((NF + 255) / 256), 256, 0, stream>>>(
      pre, N, bsc, bsh, agg, 128, 0.01f);

  // ---- out = t1 @ W2 + b2   (ncols=64 -> 4 waves/block)
  gemm_wmma_f32<<<rowTiles, 128, 0, stream>>>(
      agg, 128, W2, nullptr, 0, nullptr, b2, out, 64, 128, 64);
}